// MixProp_25993142075464
// MI455X (gfx1250) — compile-verified
//
#include <hip/hip_runtime.h>

// MixProp on MI455X (gfx1250, wave32).
// Phase 1 (3x, sequential): prop = adj @ h  as 256 batched GEMMs M=512,K=512,N=168 (fp32 WMMA)
//                           h' = 0.05*x + 0.95*prop   (fused epilogue)
//   - staging:   GLOBAL_LOAD_ASYNC_TO_LDS (ASYNCcnt), double-buffered in LDS
//   - compute:   V_WMMA_F32_16X16X4_F32, one 16x16 C-tile per wave
// Phase 2 (after each stage): out += W_g @ h_g  (bandwidth-bound channel mix, VALU + LDS W)

#define ALPHA 0.05f
#define CIN   32
#define COUT  32
#define NN    512
#define TT    168
#define BB    8

typedef __attribute__((ext_vector_type(2))) float v2f;
typedef __attribute__((ext_vector_type(8))) float v8f;

#define KC      64      // K chunk resident in LDS
#define NCHUNK  (NN / KC)
#define LDA     68      // padded LDS strides (68 mod 64 = 4 -> conflict-free column reads)
#define LDB     68
#define ASYNC_PER_CHUNK 12   // per-wave async issues per chunk: 4 x b128 (A) + 8 x b32 (B)

// -------------------------------------------------------------------------
// Propagation GEMM: one workgroup = 64(v) x 32(t) tile of one (b,c) batch.
// -------------------------------------------------------------------------
__global__ __launch_bounds__(256) void prop_wmma(const float* __restrict__ hin,
                                                 const float* __restrict__ xin,
                                                 const float* __restrict__ adj,
                                                 float* __restrict__ hout)
{
    __shared__ __align__(16) float sA [2][64 * LDA];   // adj tile  [v-local][w-local]
    __shared__ __align__(16) float sBt[2][32 * LDB];   // h tile, transposed: [t-local][w-local]

    const int tid  = threadIdx.x;
    const int lane = tid & 31;
    const int wave = tid >> 5;

    const int tb = blockIdx.x;            // t tile (32 wide, 6 tiles, last ragged)
    const int vb = blockIdx.y;            // v tile (64 tall)
    const int bc = blockIdx.z;            // b*C + c

    const float* hmat = hin  + (size_t)bc * NN * TT;
    const float* xmat = xin  + (size_t)bc * NN * TT;
    float*       omat = hout + (size_t)bc * NN * TT;

    const int v0 = vb * 64;
    const int t0 = tb * 32;

    // wave's 16x16 subtile inside the 64x32 block tile
    const int wm = (wave >> 1) * 16;      // local row   0/16/32/48
    const int wn = (wave &  1) * 16;      // local col   0/16

    // per-lane fragment addressing (ISA 7.12.2):
    //  A 16x4 : lane&15 = M, VGPR0/1 = consecutive K, lanes 16-31 at K+2
    //  B 4x16 : lane&15 = N, VGPR0/1 = consecutive K, lanes 16-31 at K+2
    const int fm = lane & 15;
    const int kq = (lane >> 4) * 2;

    // LDS byte offsets (flat shared pointers carry the LDS offset in the low 32 bits)
    const unsigned ldsA[2] = { (unsigned)(size_t)&sA [0][0], (unsigned)(size_t)&sA [1][0] };
    const unsigned ldsB[2] = { (unsigned)(size_t)&sBt[0][0], (unsigned)(size_t)&sBt[1][0] };

    const bool tok = (t0 + lane) < TT;    // ragged-t mask for B staging

    // ---- async staging of one K-chunk into LDS buffer `buf` (12 issues/wave) ----
    auto issue_chunk = [&](int k0, int buf) {
        const unsigned aBase = ldsA[buf];
        const unsigned bBase = ldsB[buf];
        // adj tile: 64 x 64 floats, 4 x b128 per thread, row-major
        #pragma unroll
        for (int i = 0; i < 4; ++i) {
            int s  = tid + 256 * i;            // 0..1023 float4 slots
            int r  = s >> 4;                   // row 0..63
            int c4 = (s & 15) << 2;            // col 0..60 step 4
            const float* g = adj + (size_t)(v0 + r) * NN + (k0 + c4);
            unsigned     l = aBase + (unsigned)((r * LDA + c4) * 4);
            asm volatile("global_load_async_to_lds_b128 %0, %1, off"
                         :: "v"(l), "v"(g) : "memory");
        }
        // h tile, transposed scatter: b32 per element, lane = t column
        #pragma unroll
        for (int i = 0; i < 8; ++i) {
            int w = wave + 8 * i;              // local w 0..63
            const float* g = tok ? (hmat + (size_t)(k0 + w) * TT + (t0 + lane)) : hmat;
            unsigned     l = bBase + (unsigned)((lane * LDB + w) * 4);
            asm volatile("global_load_async_to_lds_b32 %0, %1, off"
                         :: "v"(l), "v"(g) : "memory");
        }
    };

    v8f acc = {};

    issue_chunk(0, 0);
    for (int c = 0; c < NCHUNK; ++c) {
        if (c + 1 < NCHUNK) {
            // prefetch next chunk, then wait until only those 12 remain outstanding:
            // async loads complete in order, so the current chunk is in LDS.
            issue_chunk((c + 1) * KC, (c + 1) & 1);
            asm volatile("s_wait_asynccnt %0" :: "i"(ASYNC_PER_CHUNK) : "memory");
        } else {
            asm volatile("s_wait_asynccnt 0" ::: "memory");
        }
        __syncthreads();   // every wave has passed its own asynccnt wait -> LDS coherent

        const float* cA = &sA [c & 1][0];
        const float* cB = &sBt[c & 1][0];
        #pragma unroll
        for (int kk = 0; kk < KC; kk += 4) {
            v2f afrag = *(const v2f*)(&cA[(wm + fm) * LDA + kk + kq]);
            v2f bfrag = *(const v2f*)(&cB[(wn + fm) * LDB + kk + kq]);
            acc = __builtin_amdgcn_wmma_f32_16x16x4_f32(
                      false, afrag, false, bfrag, (short)0, acc, false, false);
        }
        __syncthreads();   // all waves done with this buffer before chunk c+2 overwrites it
    }

    // ---- epilogue: h' = alpha*x + (1-alpha)*prop, masked on ragged t ----
    const int tcol = t0 + wn + (lane & 15);
    if (tcol < TT) {
        #pragma unroll
        for (int i = 0; i < 8; ++i) {
            int vrow = v0 + wm + i + 8 * (lane >> 4);      // C layout: M = i + 8*(lane/16)
            size_t off = (size_t)vrow * TT + tcol;
            omat[off] = ALPHA * xmat[off] + (1.0f - ALPHA) * acc[i];
        }
    }
}

// -------------------------------------------------------------------------
// Channel mix: out[b,o,r] (+)= sum_c W[o, g*32+c] * h[b,c,r]   (+ bias on init)
// Pure bandwidth; one thread per spatial point, W block in LDS.
// -------------------------------------------------------------------------
__global__ __launch_bounds__(256) void mix_add(const float* __restrict__ h,
                                               const float* __restrict__ W,
                                               const float* __restrict__ bias,
                                               float* __restrict__ out,
                                               int gblk, int initFlag)
{
    __shared__ float sW[COUT * CIN];
    const int tid = threadIdx.x;
    #pragma unroll
    for (int i = tid; i < COUT * CIN; i += 256) {
        int o = i / CIN, c = i % CIN;
        sW[i] = W[o * (4 * CIN) + gblk * CIN + c];
    }
    __syncthreads();

    const size_t NT    = (size_t)NN * TT;
    const size_t total = (size_t)BB * NT;
    size_t p = (size_t)blockIdx.x * 256 + tid;
    if (p >= total) return;
    const size_t b = p / NT;
    const size_t r = p % NT;

    const float* hb = h   + b * CIN  * NT + r;
    float*       ob = out + b * COUT * NT + r;

    float accv[COUT];
    #pragma unroll
    for (int o = 0; o < COUT; ++o)
        accv[o] = initFlag ? bias[o] : ob[(size_t)o * NT];

    #pragma unroll 4
    for (int c = 0; c < CIN; ++c) {
        float hv = hb[(size_t)c * NT];
        #pragma unroll
        for (int o = 0; o < COUT; ++o)
            accv[o] += sW[o * CIN + c] * hv;
    }

    #pragma unroll
    for (int o = 0; o < COUT; ++o)
        ob[(size_t)o * NT] = accv[o];
}

// -------------------------------------------------------------------------
extern "C" void kernel_launch(void* const* d_in, const int* in_sizes, int n_in,
                              void* d_out, int out_size, void* d_ws, size_t ws_size,
                              hipStream_t stream)
{
    const float* x    = (const float*)d_in[0];   // [8,32,512,168]
    const float* adj  = (const float*)d_in[1];   // [512,512]
    const float* W    = (const float*)d_in[2];   // [32,128]
    const float* bias = (const float*)d_in[3];   // [32]
    float*       out  = (float*)d_out;           // [8,32,512,168]

    const size_t per = (size_t)BB * CIN * NN * TT;   // elements per h stage
    float* hA = (float*)d_ws;                        // ping
    float* hB = hA + per;                            // pong (needs 2*per*4 = 176 MB ws)

    dim3 blk(256);
    dim3 pgrid((TT + 31) / 32, NN / 64, BB * CIN);   // 6 x 8 x 256 = 12288 WGs
    dim3 mgrid((unsigned)((per / CIN + 255) / 256)); // B*N*T points

    // stage 0: out = bias + W0 @ x
    mix_add  <<<mgrid, blk, 0, stream>>>(x,  W, bias, out, 0, 1);
    // h1 = 0.05 x + 0.95 adj@x ; out += W1 @ h1
    prop_wmma<<<pgrid, blk, 0, stream>>>(x,  x, adj, hA);
    mix_add  <<<mgrid, blk, 0, stream>>>(hA, W, bias, out, 1, 0);
    // h2 ; out += W2 @ h2
    prop_wmma<<<pgrid, blk, 0, stream>>>(hA, x, adj, hB);
    mix_add  <<<mgrid, blk, 0, stream>>>(hB, W, bias, out, 2, 0);
    // h3 ; out += W3 @ h3
    prop_wmma<<<pgrid, blk, 0, stream>>>(hB, x, adj, hA);
    mix_add  <<<mgrid, blk, 0, stream>>>(hA, W, bias, out, 3, 0);
}